// LoRALinear_9732395893261
// MI455X (gfx1250) — compile-verified
//
#include <hip/hip_runtime.h>
#include <hip/hip_bf16.h>
#include <stddef.h>
#include <stdint.h>

// ---------------------------------------------------------------------------
// LoRA linear for MI455X (gfx1250, wave32):
//   out = x @ W^T + bias + scale[b] * (x @ A[ad]^T) @ Bm[ad]^T
// Base GEMM on native fp32 WMMA (v_wmma_f32_16x16x4_f32), with CDNA5
// async global->LDS copies (ASYNCcnt) and double-buffered LDS staging.
// ---------------------------------------------------------------------------

typedef float  v2f __attribute__((ext_vector_type(2)));
typedef float  v8f __attribute__((ext_vector_type(8)));

#define D_IN   4096
#define D_OUT  4096
#define BATCH  4
#define SEQ    2048
#define RANK   16
#define ROWS   (BATCH * SEQ)      // 8192

#define TM 64                     // workgroup M tile
#define TN 128                    // workgroup N tile
#define TK 32                     // K stage depth
#define LDP (TK + 4)              // padded LDS row stride: 36 floats = 144 B
                                  //  -> 16B-aligned B128 async writes AND
                                  //     conflict-free column reads (36*m mod 64 distinct)
#define NSTAGES (D_IN / TK)       // 128

// ---------------------------------------------------------------------------
// CDNA5 async global->LDS copy (GVS mode), tracked by ASYNCcnt.
//   LDS byte address in a VGPR; global address = SGPR base + VGPR byte offset
// ---------------------------------------------------------------------------
__device__ __forceinline__ void async_copy_b128(const float* base,
                                                unsigned gbyteoff,
                                                void* lds_dst) {
    unsigned loff = (unsigned)(uintptr_t)lds_dst;   // low 32 bits = LDS offset
    asm volatile("global_load_async_to_lds_b128 %0, %1, %2"
                 :
                 : "v"(loff), "v"(gbyteoff), "s"(base)
                 : "memory");
}

__device__ __forceinline__ void wait_async_zero() {
    asm volatile("s_wait_asynccnt 0" ::: "memory");
}

// ---------------------------------------------------------------------------
// Kernel 1: low_scaled[row, r] = scale[adapter] * sum_i x[row,i] * A[adapter,r,i]
// One block (256 threads = 8 waves) per row; wave w covers ranks w and w+8.
// ---------------------------------------------------------------------------
__global__ __launch_bounds__(256)
void lora_low_kernel(const float* __restrict__ x,
                     const float* __restrict__ lora_a,     // [8, 16, 4096]
                     const float* __restrict__ scaling,    // [8]
                     const int*   __restrict__ adapter_id, // [4]
                     float*       __restrict__ low)        // [8192, 16]
{
    const int row  = blockIdx.x;
    const int b    = row >> 11;               // row / 2048
    const int ad   = adapter_id[b];
    const float sc = scaling[ad];

    const float* __restrict__ xr  = x + (size_t)row * D_IN;
    const float* __restrict__ Aad = lora_a + (size_t)ad * RANK * D_IN;

    const int wave = threadIdx.x >> 5;
    const int lane = threadIdx.x & 31;

    #pragma unroll
    for (int rr = 0; rr < 2; ++rr) {
        const int r = wave + rr * 8;
        const float* __restrict__ Ar = Aad + (size_t)r * D_IN;
        float sum = 0.0f;
        #pragma unroll 4
        for (int i = lane; i < D_IN; i += 32)
            sum += xr[i] * Ar[i];
        #pragma unroll
        for (int off = 16; off > 0; off >>= 1)
            sum += __shfl_down(sum, off, 32);
        if (lane == 0)
            low[(size_t)row * RANK + r] = sc * sum;
    }
}

// ---------------------------------------------------------------------------
// Kernel 2: double-buffered async-LDS fp32 WMMA GEMM with fused LoRA + bias.
// Block = 256 threads = 8 waves in 4(M) x 2(N) grid; wave tile 16 x 64.
// ---------------------------------------------------------------------------
__global__ __launch_bounds__(256)
void lora_gemm_kernel(const float* __restrict__ x,          // [8192, 4096]
                      const float* __restrict__ W,          // [4096, 4096]  [Dout, Din]
                      const float* __restrict__ bias,       // [4096]
                      const float* __restrict__ lora_b,     // [8, 4096, 16]
                      const int*   __restrict__ adapter_id, // [4]
                      const float* __restrict__ low,        // [8192, 16] (scale folded in)
                      float*       __restrict__ out)        // [8192, 4096]
{
    __shared__ float lx[2][TM * LDP];   // x tile   64 x 36, double buffered
    __shared__ float lw[2][TN * LDP];   // W tile  128 x 36, double buffered

    const int row0 = blockIdx.y * TM;
    const int col0 = blockIdx.x * TN;
    const int tid  = threadIdx.x;
    const int wave = tid >> 5;
    const int lane = tid & 31;
    const int wm   = wave & 3;           // wave M index (0..3)
    const int wn   = wave >> 2;          // wave N index (0..1)

    // fp32 WMMA lane mapping (ISA 7.12.2):
    //   A 16x4:  m = lane%16, k = 2*(lane/16) + {0,1}
    //   B 4x16:  n = lane%16, k = 2*(lane/16) + {0,1}
    //   C/D:     n = lane%16, M = vgpr + 8*(lane/16)
    const int m16 = lane & 15;
    const int kb  = (lane >> 4) << 1;    // 0 or 2

    // Per-thread async-copy assignments (same every stage).
    const int xidx0 = tid,        xidx1 = tid + 256;     // 512 x-chunks of 4 floats
    const int xr0 = xidx0 >> 3,   xc0 = (xidx0 & 7) << 2;
    const int xr1 = xidx1 >> 3,   xc1 = (xidx1 & 7) << 2;

    v8f acc[4] = {v8f{}, v8f{}, v8f{}, v8f{}};

    // ---- Prologue: issue async loads for stage 0 into buffer 0 -------------
    {
        const int k0 = 0;
        async_copy_b128(x, (unsigned)(((row0 + xr0) * D_IN + k0 + xc0) * 4), &lx[0][xr0 * LDP + xc0]);
        async_copy_b128(x, (unsigned)(((row0 + xr1) * D_IN + k0 + xc1) * 4), &lx[0][xr1 * LDP + xc1]);
        #pragma unroll
        for (int i = 0; i < 4; ++i) {
            const int idx = tid + i * 256;
            const int r   = idx >> 3;
            const int c4  = (idx & 7) << 2;
            async_copy_b128(W, (unsigned)(((col0 + r) * D_IN + k0 + c4) * 4), &lw[0][r * LDP + c4]);
        }
    }

    for (int s = 0; s < NSTAGES; ++s) {
        const int cur = s & 1;
        const int nxt = cur ^ 1;

        wait_async_zero();      // my async writes for buf[cur] have landed
        __syncthreads();        // everyone's have; buf[nxt] no longer being read

        // ---- Issue async loads for stage s+1 into buf[nxt] ----------------
        if (s + 1 < NSTAGES) {
            const int k0 = (s + 1) * TK;
            async_copy_b128(x, (unsigned)(((row0 + xr0) * D_IN + k0 + xc0) * 4), &lx[nxt][xr0 * LDP + xc0]);
            async_copy_b128(x, (unsigned)(((row0 + xr1) * D_IN + k0 + xc1) * 4), &lx[nxt][xr1 * LDP + xc1]);
            #pragma unroll
            for (int i = 0; i < 4; ++i) {
                const int idx = tid + i * 256;
                const int r   = idx >> 3;
                const int c4  = (idx & 7) << 2;
                async_copy_b128(W, (unsigned)(((col0 + r) * D_IN + k0 + c4) * 4), &lw[nxt][r * LDP + c4]);
            }
        }

        // ---- Compute stage s from buf[cur]: 8 k-steps x 4 WMMA -------------
        #pragma unroll
        for (int kk = 0; kk < TK; kk += 4) {
            v2f a;
            a.x = lx[cur][(wm * 16 + m16) * LDP + kk + kb];
            a.y = lx[cur][(wm * 16 + m16) * LDP + kk + kb + 1];
            #pragma unroll
            for (int nb = 0; nb < 4; ++nb) {
                const int n = wn * 64 + nb * 16 + m16;
                v2f bf;
                bf.x = lw[cur][n * LDP + kk + kb];
                bf.y = lw[cur][n * LDP + kk + kb + 1];
                acc[nb] = __builtin_amdgcn_wmma_f32_16x16x4_f32(
                    /*neg_a=*/false, a, /*neg_b=*/false, bf,
                    /*c_mod=*/(short)0, acc[nb],
                    /*reuse_a=*/false, /*reuse_b=*/false);
            }
        }
    }

    // ---- Fused LoRA delta: acc += low_scaled(16xR) @ lora_b(n, R)^T --------
    // 64-row tiles never cross a batch boundary (2048 % 64 == 0).
    const int b  = row0 >> 11;
    const int ad = adapter_id[b];
    const float* __restrict__ Bad = lora_b + (size_t)ad * D_OUT * RANK;
    const int wrow  = row0 + wm * 16;
    const int wcol0 = col0 + wn * 64;

    #pragma unroll
    for (int rk = 0; rk < RANK; rk += 4) {
        v2f a;
        a.x = low[(size_t)(wrow + m16) * RANK + rk + kb];
        a.y = low[(size_t)(wrow + m16) * RANK + rk + kb + 1];
        #pragma unroll
        for (int nb = 0; nb < 4; ++nb) {
            const int n = wcol0 + nb * 16 + m16;
            v2f bf;
            bf.x = Bad[(size_t)n * RANK + rk + kb];
            bf.y = Bad[(size_t)n * RANK + rk + kb + 1];
            acc[nb] = __builtin_amdgcn_wmma_f32_16x16x4_f32(
                false, a, false, bf, (short)0, acc[nb], false, false);
        }
    }

    // ---- Bias + store ------------------------------------------------------
    const int mhalf = (lane >> 4) << 3;   // +0 or +8 rows
    #pragma unroll
    for (int nb = 0; nb < 4; ++nb) {
        const int n  = wcol0 + nb * 16 + m16;
        const float bv = bias[n];
        #pragma unroll
        for (int v = 0; v < 8; ++v) {
            const int r = wrow + v + mhalf;
            out[(size_t)r * D_OUT + n] = acc[nb][v] + bv;
        }
    }
}

// ---------------------------------------------------------------------------
// Harness entry point.
// Input order: x, weight, bias, lora_a_weights, lora_b_weights,
//              scaling_factors, adapter_id
// ---------------------------------------------------------------------------
extern "C" void kernel_launch(void* const* d_in, const int* in_sizes, int n_in,
                              void* d_out, int out_size, void* d_ws, size_t ws_size,
                              hipStream_t stream) {
    const float* x       = (const float*)d_in[0];
    const float* weight  = (const float*)d_in[1];
    const float* bias    = (const float*)d_in[2];
    const float* lora_a  = (const float*)d_in[3];
    const float* lora_b  = (const float*)d_in[4];
    const float* scaling = (const float*)d_in[5];
    const int*   adid    = (const int*)d_in[6];
    float*       out     = (float*)d_out;
    float*       low     = (float*)d_ws;   // [8192, 16] fp32 scratch = 512 KB

    // Kernel 1: rank-16 projection with scale folded in.
    lora_low_kernel<<<ROWS, 256, 0, stream>>>(x, lora_a, scaling, adid, low);

    // Kernel 2: async double-buffered WMMA GEMM + fused LoRA delta + bias.
    dim3 grid(D_OUT / TN, ROWS / TM);   // (32, 128)
    lora_gemm_kernel<<<grid, 256, 0, stream>>>(x, weight, bias, lora_b, adid, low, out);
}